// DarcyLoss_19713899889304
// MI455X (gfx1250) — compile-verified
//
#include <hip/hip_runtime.h>
#include <hip/hip_bf16.h>
#include <stdint.h>

// MI455X / gfx1250. Memory-bound (~402 MiB @ 23.3 TB/s ~= 17us floor).
// FD stencils = f32 WMMA (16x16x4) matmuls against constant banded operators
// held in LDS; x0_hat planes staged via global_load_async_to_lds_b128.
// Axis-0 derivatives computed transposed (p^T G) so only B-side operator
// fragments are needed and all boundary terms become lane-uniform weights.

typedef __attribute__((ext_vector_type(2))) float v2f;
typedef __attribute__((ext_vector_type(8))) float v8f;

#define HH 64
#define WWID 64
#define NB 4096
#define NTOT (NB * 2 * HH * WWID)      // 33554432
#define MSE_BLOCKS 2048
#define SMEM_FLOATS (4 * HH * WWID + 128)        // P,Q,G1,G2 + reduction
#define SMEM_BYTES  (SMEM_FLOATS * 4)            // 66048 B (dynamic LDS)

__device__ __forceinline__ v8f wmma4(v2f a, v2f b, v8f c) {
  // D = A(16x4,f32) x B(4x16,f32) + C(16x16,f32)
  return __builtin_amdgcn_wmma_f32_16x16x4_f32(false, a, false, b, (short)0, c,
                                               false, false);
}

// ================= Kernel 1: streaming MSE partial sums =================
__global__ __launch_bounds__(256) void mse_partial_kernel(
    const float* __restrict__ a, const float* __restrict__ b,
    float* __restrict__ partial, int n4) {
  const int tid = threadIdx.x;
  const int stride = gridDim.x * blockDim.x;
  const float4* a4 = (const float4*)a;
  const float4* b4 = (const float4*)b;
  float s = 0.0f;
  for (int i = blockIdx.x * blockDim.x + tid; i < n4; i += stride) {
    if (i + stride < n4) {
      __builtin_prefetch(&a4[i + stride], 0, 1);   // global_prefetch_b8
      __builtin_prefetch(&b4[i + stride], 0, 1);
    }
    float4 x = a4[i];
    float4 y = b4[i];
    float d0 = x.x - y.x, d1 = x.y - y.y, d2 = x.z - y.z, d3 = x.w - y.w;
    s += d0 * d0 + d1 * d1 + d2 * d2 + d3 * d3;
  }
  __shared__ float red[256];
  red[tid] = s;
  __syncthreads();
  for (int off = 128; off > 0; off >>= 1) {   // fixed pairing -> deterministic
    if (tid < off) red[tid] += red[tid + off];
    __syncthreads();
  }
  if (tid == 0) partial[blockIdx.x] = red[0];
}

// ================= Kernel 2: per-batch Darcy residual (WMMA) =============
__global__ __launch_bounds__(128) void residual_kernel(
    const float* __restrict__ x0, const float* __restrict__ var,
    float* __restrict__ contrib) {
  extern __shared__ float smem[];
  float* sP   = smem;                         // 4096: pressure plane
  float* sQ   = smem + HH * WWID;             // 4096: permeability plane
  float* sG1  = smem + 2 * HH * WWID;         // 4096: 1st-deriv operator
  float* sG2  = smem + 3 * HH * WWID;         // 4096: 2nd-deriv operator
  float* sRed = smem + 4 * HH * WWID;         // 128

  const int tid = threadIdx.x;
  const int b = blockIdx.x;
  const float* gp = x0 + (size_t)b * (2 * HH * WWID);
  const float* gq = gp + HH * WWID;

  // ---- async global -> LDS staging (16 KiB per plane, b128 per lane) ----
  unsigned ldsP = (unsigned)(uintptr_t)(void*)sP;  // flat LDS addr low 32b
  unsigned ldsQ = (unsigned)(uintptr_t)(void*)sQ;
  uint64_t baseP = (uint64_t)(uintptr_t)gp;
  uint64_t baseQ = (uint64_t)(uintptr_t)gq;
#pragma unroll
  for (int it = 0; it < 8; ++it) {
    unsigned off = (unsigned)(tid + it * 128) * 16u;
    asm volatile("global_load_async_to_lds_b128 %0, %1, %2"
                 :: "v"(ldsP + off), "v"(off), "s"(baseP) : "memory");
    asm volatile("global_load_async_to_lds_b128 %0, %1, %2"
                 :: "v"(ldsQ + off), "v"(off), "s"(baseQ) : "memory");
  }

  // ---- build G1/G2 in LDS (overlapped with async loads) ----
  // zero-fill 32 KiB with b128 stores
  {
    float4 z4 = make_float4(0.f, 0.f, 0.f, 0.f);
    float4* g4 = (float4*)sG1;                // sG1,sG2 contiguous: 2048 f4
#pragma unroll
    for (int i = 0; i < 16; ++i) g4[tid + i * 128] = z4;
  }
  __syncthreads();
  // branchless tap scatter: threads 0..63 -> G1 column j, 64..127 -> G2 col j
  {
    const int j = tid & 63;
    const bool lo = (j == 0), hi = (j == 63);
    if (tid < 64) {
      const float c = 31.5f;                  // 1/(2D), D = 1/63
      int   kd = lo ? 2 : (hi ? 63 : j);
      float wd = lo ? -c : (hi ? 3.0f * c : 0.0f);
      int   ka = lo ? 0 : (hi ? 61 : j - 1);
      float wa = lo ? -3.0f * c : (hi ? c : -c);
      int   kb = lo ? 1 : (hi ? 62 : j + 1);
      float wb = lo ? 4.0f * c : (hi ? -4.0f * c : c);
      sG1[kd * WWID + j] = wd;                // dummy first (in-order LDS)
      sG1[ka * WWID + j] = wa;
      sG1[kb * WWID + j] = wb;
    } else {
      const float s = 3969.0f;                // 1/D^2
      int   kd = lo ? 3 : (hi ? 63 : j);
      float wd = lo ? -s : (hi ? 2.0f * s : 0.0f);
      int   ka = lo ? 0 : (hi ? 60 : j - 1);
      float wa = lo ? 2.0f * s : (hi ? -s : s);
      int   kb = lo ? 1 : (hi ? 61 : j);
      float wb = lo ? -5.0f * s : (hi ? 4.0f * s : -2.0f * s);
      int   kc = lo ? 2 : (hi ? 62 : j + 1);
      float wc = lo ? 4.0f * s : (hi ? -5.0f * s : s);
      sG2[kd * WWID + j] = wd;                // dummy first, kb overwrites
      sG2[ka * WWID + j] = wa;
      sG2[kb * WWID + j] = wb;
      sG2[kc * WWID + j] = wc;
    }
  }
  asm volatile("s_wait_asynccnt 0" ::: "memory");
  __syncthreads();

  // ---- per-wave 16-row band; 4 column tiles; 6 fused matmuls ----
  const int wave = tid >> 5;
  const int lane = tid & 31;
  const int l = lane & 15;
  const int hi2 = lane >> 4;         // 0: K {0,1} ; 1: K {2,3} (rows +8 for C)
  const int m0 = wave << 4;
  const int mrow = m0 + l;           // A-fragment row for this lane
  float acc = 0.0f;

  for (int nt = 0; nt < 4; ++nt) {
    const int n0 = nt << 4;
    const int ncol = n0 + l;
    // boundary-condition weight (all BC terms are column conditions here)
    const float w1 = ((ncol == 0) ? 1.0f : 0.0f) -
                     ((ncol == WWID - 1) ? 1.0f : 0.0f);
    v8f c_pd1   = {0.f, 0.f, 0.f, 0.f, 0.f, 0.f, 0.f, 0.f};
    v8f c_pd11  = c_pd1, c_qd1 = c_pd1;
    v8f c_pd0t  = c_pd1, c_pd00t = c_pd1, c_qd0t = c_pd1;
#pragma unroll 4
    for (int k0 = 0; k0 < 64; k0 += 4) {
      const int kk = k0 + (hi2 ? 2 : 0);
      v2f aP, aQ, aPt, aQt, bG1, bG2;
      aP.x  = sP[mrow * WWID + kk];      aP.y  = sP[mrow * WWID + kk + 1];
      aQ.x  = sQ[mrow * WWID + kk];      aQ.y  = sQ[mrow * WWID + kk + 1];
      aPt.x = sP[kk * WWID + mrow];      aPt.y = sP[(kk + 1) * WWID + mrow];
      aQt.x = sQ[kk * WWID + mrow];      aQt.y = sQ[(kk + 1) * WWID + mrow];
      bG1.x = sG1[kk * WWID + ncol];     bG1.y = sG1[(kk + 1) * WWID + ncol];
      bG2.x = sG2[kk * WWID + ncol];     bG2.y = sG2[(kk + 1) * WWID + ncol];
      c_pd1   = wmma4(aP,  bG1, c_pd1);    // p_d1   = p G1
      c_pd11  = wmma4(aP,  bG2, c_pd11);   // p_d11  = p G2
      c_qd1   = wmma4(aQ,  bG1, c_qd1);    // perm_d1
      c_pd0t  = wmma4(aPt, bG1, c_pd0t);   // p_d0^T  = p^T G1
      c_pd00t = wmma4(aPt, bG2, c_pd00t);  // p_d00^T = p^T G2
      c_qd0t  = wmma4(aQt, bG1, c_qd0t);   // perm_d0^T
    }
    // elementwise eq0 + BC; only the global sum is needed, so transposed
    // pairs contribute directly.
#pragma unroll
    for (int r = 0; r < 8; ++r) {
      const int row = m0 + r + (hi2 << 3);
      const float qq1 = sQ[row * WWID + ncol];   // perm[i=row][j=ncol]
      const float qq2 = sQ[ncol * WWID + row];   // perm[i=ncol][j=row]
      const float fs = (row < 8 && ncol < 8) ? 10.0f
                     : ((row >= 56 && ncol >= 56) ? -10.0f : 0.0f);
      const float pd1 = c_pd1[r], pd11 = c_pd11[r], qd1 = c_qd1[r];
      const float pd0 = c_pd0t[r], pd00 = c_pd00t[r], qd0 = c_qd0t[r];
      acc += -qq1 * pd11 - qd1 * pd1 + w1 * pd1 - fs
             - qq2 * pd00 - qd0 * pd0 - w1 * pd0;
    }
  }

  sRed[tid] = acc;
  __syncthreads();
  if (tid == 0) {
    float s = 0.0f;
#pragma unroll 1
    for (int t = 0; t < 128; ++t) s += sRed[t];     // fixed order
    const float r = s * (1.0f / (float)(HH * WWID * 3));
    float c = 0.5f * r * r / var[b];                // -ll before clamp
    contrib[b] = fminf(c, 27.6310211159f);          // -max(ll, CLAMP_MIN)
  }
}

// ================= Kernel 3: deterministic finalize =================
__global__ void finalize_kernel(const float* __restrict__ partial, int nPart,
                                const float* __restrict__ contrib, int nB,
                                float* __restrict__ out) {
  if (threadIdx.x == 0 && blockIdx.x == 0) {
    float s = 0.0f;
    for (int i = 0; i < nPart; ++i) s += partial[i];
    const float data_loss = s / (float)NTOT;
    float rs = 0.0f;
    for (int i = 0; i < nB; ++i) rs += contrib[i];
    out[0] = data_loss + rs / (float)nB;
  }
}

extern "C" void kernel_launch(void* const* d_in, const int* in_sizes, int n_in,
                              void* d_out, int out_size, void* d_ws,
                              size_t ws_size, hipStream_t stream) {
  (void)in_sizes; (void)n_in; (void)out_size; (void)ws_size;
  const float* model_out = (const float*)d_in[0];
  const float* target    = (const float*)d_in[1];
  const float* x0_hat    = (const float*)d_in[2];
  const float* var       = (const float*)d_in[3];
  float* out = (float*)d_out;

  float* wsPartial = (float*)d_ws;            // MSE_BLOCKS floats
  float* wsContrib = wsPartial + MSE_BLOCKS;  // NB floats

  const int n4 = NTOT / 4;
  mse_partial_kernel<<<MSE_BLOCKS, 256, 0, stream>>>(model_out, target,
                                                     wsPartial, n4);
  residual_kernel<<<NB, 128, SMEM_BYTES, stream>>>(x0_hat, var, wsContrib);
  finalize_kernel<<<1, 32, 0, stream>>>(wsPartial, MSE_BLOCKS, wsContrib, NB,
                                        out);
}